// GeoFeat_15496242004733
// MI455X (gfx1250) — compile-verified
//
#include <hip/hip_runtime.h>
#include <cstddef>

// GeoFeat edge-MLP for MI455X (gfx1250), fp32 throughout via V_WMMA_F32_16X16X4_F32.
// N=20000 nodes, E=320000 edges, D=128, A=32, HDIM=441 (padded to 448).
// Round 2: register-block MT output tiles along M per wave -> B fragments reused
// MT times per load, MT independent WMMA accumulation chains.

#define D_DIM 128
#define HDIM_  441
#define HPAD  448
#define BN_EPS 1e-5f

typedef __attribute__((ext_vector_type(2))) float v2f;
typedef __attribute__((ext_vector_type(8))) float v8f;

__device__ __forceinline__ v8f wmma4(v2f a, v2f b, v8f c) {
  // D = A(16x4) * B(4x16) + C(16x16), full fp32
  return __builtin_amdgcn_wmma_f32_16x16x4_f32(false, a, false, b, (short)0, c, false, false);
}

// One wave computes MT stacked 16x16 output tiles (rows m0 + t*16 .. +15, cols n0..n0+15).
// A-stripes come from LDS (ds_load_b64 pairs), B (weights) from global (L2-hot),
// each B fragment reused across the MT accumulators.
// Layouts per CDNA5 ISA 7.12.2:
//   A 16x4 f32: lane<16 -> {K=k0,k0+1}, lane>=16 -> {K=k0+2,k0+3}, M = lane%16
//   B 4x16 f32: VGPR r holds row K = k0 + r + 2*(lane>=16), N = lane%16
//   C/D 16x16:  VGPR r holds row M = r + 8*(lane>=16),      N = lane%16
template<int K, int LDW, int MT>
__device__ __forceinline__ void wmma_tile(const float* __restrict__ As,
                                          const float* __restrict__ W,
                                          const float* __restrict__ bias,
                                          float* __restrict__ Y, int ldy, int m0) {
  const int tid  = threadIdx.x;
  const int wave = tid >> 5;
  const int lane = tid & 31;
  const int half = lane >> 4;
  const int ln   = lane & 15;
  const int n0   = wave * 16;

  v8f acc[MT];
#pragma unroll
  for (int t = 0; t < MT; ++t)
    acc[t] = (v8f){0.f, 0.f, 0.f, 0.f, 0.f, 0.f, 0.f, 0.f};

  const float* Wc = W + n0 + ln;

#pragma unroll 4
  for (int k0 = 0; k0 < K; k0 += 4) {
    const int kk = k0 + half * 2;
    v2f b;
    b.x = Wc[(size_t)kk * LDW];
    b.y = Wc[(size_t)(kk + 1) * LDW];
#pragma unroll
    for (int t = 0; t < MT; ++t) {
      const float* Arow = As + (size_t)(t * 16 + ln) * K;
      v2f a;
      a.x = Arow[kk];
      a.y = Arow[kk + 1];
      acc[t] = wmma4(a, b, acc[t]);
    }
  }

  const float bv = bias[n0 + ln];
#pragma unroll
  for (int t = 0; t < MT; ++t) {
#pragma unroll
    for (int r = 0; r < 8; ++r) {
      Y[(size_t)(m0 + t * 16 + r + half * 8) * ldy + (n0 + ln)] = acc[t][r] + bv;
    }
  }
}

// ---- projections: Y[rows,96] = A[rows,128] @ W[128,96] + b ----  (32 rows/block)
__global__ __launch_bounds__(192) void k_proj(const float* __restrict__ A,
                                              const float* __restrict__ W,
                                              const float* __restrict__ bias,
                                              float* __restrict__ Y) {
  __shared__ float As[32 * 128];
  const int m0 = blockIdx.x * 32;
  for (int i = threadIdx.x; i < 32 * 128; i += 192)
    As[i] = A[(size_t)m0 * 128 + i];
  __syncthreads();
  wmma_tile<128, 96, 2>(As, W, bias, Y, 96, m0);
}

// ---- geometry: build H[E,448] ----
// layout per row: Ks[0:192] Qt[192:384] quat[384:393] rbf[393:409] QRK[409:441] pad[441:448]
__global__ __launch_bounds__(256) void k_geom(const float* __restrict__ Vedge,
                                              const float* __restrict__ Vloc,
                                              const float* __restrict__ rot,
                                              const float* __restrict__ trans,
                                              const float* __restrict__ rbf,
                                              const long long* __restrict__ eidx,
                                              float* __restrict__ H, int E_) {
  const int gid = blockIdx.x * blockDim.x + threadIdx.x;
  const int e = gid >> 6;
  const int j = gid & 63;
  if (e >= E_) return;

  const long long s = eidx[e];        // src
  const long long d = eidx[E_ + e];   // dst
  const float* R = rot + (size_t)e * 9;

  float kx, ky, kz;
  if (j < 32) {
    const float* p = Vedge + (size_t)e * 96 + j * 3;
    kx = p[0]; ky = p[1]; kz = p[2];
  } else {
    const float* p = Vloc + (size_t)s * 96 + (j - 32) * 3;
    kx = p[0]; ky = p[1]; kz = p[2];
  }

  float* h = H + (size_t)e * HPAD;
  h[3 * j + 0] = kx; h[3 * j + 1] = ky; h[3 * j + 2] = kz;

  const float r0 = R[0], r1 = R[1], r2 = R[2];
  const float r3 = R[3], r4 = R[4], r5 = R[5];
  const float r6 = R[6], r7 = R[7], r8 = R[8];
  const float qx = r0 * kx + r1 * ky + r2 * kz;   // (R*k): also RKs for j>=32
  const float qy = r3 * kx + r4 * ky + r5 * kz;
  const float qz = r6 * kx + r7 * ky + r8 * kz;

  h[192 + 3 * j + 0] = qx + trans[(size_t)e * 3 + 0];
  h[192 + 3 * j + 1] = qy + trans[(size_t)e * 3 + 1];
  h[192 + 3 * j + 2] = qz + trans[(size_t)e * 3 + 2];

  if (j >= 32) {  // QRK[a] = Vd[a] . (R*Vs[a])
    const int a = j - 32;
    const float* pd = Vloc + (size_t)d * 96 + a * 3;
    h[409 + a] = pd[0] * qx + pd[1] * qy + pd[2] * qz;
  }
  if (j < 9)  h[384 + j] = R[j];
  if (j < 16) h[393 + j] = rbf[(size_t)e * 16 + j];
  if (j < 7)  h[441 + j] = 0.0f;
}

// ---- GEMM1: Y1 = H @ Wp(448x128) + b1 ----  (32 rows/block, 57KB LDS)
__global__ __launch_bounds__(256) void k_gemm1(const float* __restrict__ H,
                                               const float* __restrict__ Wp,
                                               const float* __restrict__ bias,
                                               float* __restrict__ Y) {
  __shared__ float As[32 * HPAD];
  const int m0 = blockIdx.x * 32;
  for (int i = threadIdx.x; i < 32 * HPAD; i += 256)
    As[i] = H[(size_t)m0 * HPAD + i];
  __syncthreads();
  wmma_tile<HPAD, 128, 2>(As, Wp, bias, Y, 128, m0);
}

// ---- GEMM with fused BN(affine)+ReLU on the A-tile load ----  (64 rows/block)
__global__ __launch_bounds__(256) void k_gemm_bn(const float* __restrict__ Yin,
                                                 const float* __restrict__ scale,
                                                 const float* __restrict__ shift,
                                                 const float* __restrict__ W,
                                                 const float* __restrict__ bias,
                                                 float* __restrict__ Yout) {
  __shared__ float As[64 * 128];
  const int m0 = blockIdx.x * 64;
  for (int i = threadIdx.x; i < 64 * 128; i += 256) {
    const int k = i & 127;
    const float v = Yin[(size_t)m0 * 128 + i];
    As[i] = fmaxf(fmaf(v, scale[k], shift[k]), 0.0f);
  }
  __syncthreads();
  wmma_tile<128, 128, 4>(As, W, bias, Yout, 128, m0);
}

// ---- GEMM4: out = concat(h_E, G_e) @ W_m(256x128) + b_m ----  (32 rows/block)
__global__ __launch_bounds__(256) void k_gemm4(const float* __restrict__ hE,
                                               const float* __restrict__ Ge,
                                               const float* __restrict__ W,
                                               const float* __restrict__ bias,
                                               float* __restrict__ Out) {
  __shared__ float As[32 * 256];
  const int m0 = blockIdx.x * 32;
  for (int i = threadIdx.x; i < 32 * 256; i += 256) {
    const int row = i >> 8, k = i & 255;
    As[i] = (k < 128) ? hE[(size_t)(m0 + row) * 128 + k]
                      : Ge[(size_t)(m0 + row) * 128 + (k - 128)];
  }
  __syncthreads();
  wmma_tile<256, 128, 2>(As, W, bias, Out, 128, m0);
}

// ---- per-column sum / sumsq over [E,128] ----
__global__ __launch_bounds__(256) void k_stats(const float* __restrict__ Y,
                                               float* __restrict__ sums,
                                               float* __restrict__ sqs,
                                               int rowsPerBlock) {
  __shared__ float red[256];
  const int tid = threadIdx.x;
  const int col = tid & 127;
  const int half = tid >> 7;
  const int r0 = blockIdx.x * rowsPerBlock + half;
  float s = 0.f, sq = 0.f;
  for (int i = 0; i < rowsPerBlock; i += 2) {
    const float v = Y[(size_t)(r0 + i) * 128 + col];
    s += v; sq += v * v;
  }
  red[tid] = s;
  __syncthreads();
  const float st = (tid < 128) ? red[tid] + red[tid + 128] : 0.f;
  __syncthreads();
  red[tid] = sq;
  __syncthreads();
  if (tid < 128) {
    const float sqt = red[tid] + red[tid + 128];
    atomicAdd(&sums[col], st);
    atomicAdd(&sqs[col], sqt);
  }
}

__global__ void k_scale_shift(const float* __restrict__ sums, const float* __restrict__ sqs,
                              const float* __restrict__ g, const float* __restrict__ be,
                              float* __restrict__ scale, float* __restrict__ shift, float invE) {
  const int t = threadIdx.x;
  if (t < 128) {
    const float m = sums[t] * invE;
    const float v = sqs[t] * invE - m * m;
    const float rs = rsqrtf(v + BN_EPS);
    const float sc = g[t] * rs;
    scale[t] = sc;
    shift[t] = be[t] - m * sc;
  }
}

__global__ void k_zero(float* __restrict__ p, int n) {
  const int i = blockIdx.x * blockDim.x + threadIdx.x;
  if (i < n) p[i] = 0.0f;
}

__global__ void k_padw1(const float* __restrict__ W1, float* __restrict__ Wp) {
  const int i = blockIdx.x * blockDim.x + threadIdx.x;
  if (i < HPAD * D_DIM) Wp[i] = (i < HDIM_ * D_DIM) ? W1[i] : 0.0f;
}

extern "C" void kernel_launch(void* const* d_in, const int* in_sizes, int n_in,
                              void* d_out, int out_size, void* d_ws, size_t ws_size,
                              hipStream_t stream) {
  const float* h_V   = (const float*)d_in[0];
  const float* h_E   = (const float*)d_in[1];
  const float* rot   = (const float*)d_in[2];
  const float* trans = (const float*)d_in[3];
  const float* rbf   = (const float*)d_in[4];
  // d_in[5] = h_E_0: unused by the reference
  const float* W_va  = (const float*)d_in[6];
  const float* b_va  = (const float*)d_in[7];
  const float* W_vd  = (const float*)d_in[8];
  const float* b_vd  = (const float*)d_in[9];
  const float* W1    = (const float*)d_in[10];
  const float* b1    = (const float*)d_in[11];
  const float* g1    = (const float*)d_in[12];
  const float* be1   = (const float*)d_in[13];
  const float* W2    = (const float*)d_in[14];
  const float* b2    = (const float*)d_in[15];
  const float* g2    = (const float*)d_in[16];
  const float* be2   = (const float*)d_in[17];
  const float* W3    = (const float*)d_in[18];
  const float* b3    = (const float*)d_in[19];
  const float* W_m   = (const float*)d_in[20];
  const float* b_m   = (const float*)d_in[21];
  const long long* eidx = (const long long*)d_in[22];  // int64 per reference

  const int N = in_sizes[0] / D_DIM;   // 20000
  const int E = in_sizes[1] / D_DIM;   // 320000

  float* ws = (float*)d_ws;
  size_t off = 0;
  float* H  = ws + off; off += (size_t)E * HPAD;    // 573 MB
  float* Y1 = ws + off; off += (size_t)E * D_DIM;   // 164 MB
  float* Y2 = ws + off; off += (size_t)E * D_DIM;   // 164 MB
  float* Ve = ws + off; off += (size_t)E * 96;      // 123 MB
  float* Vl = ws + off; off += (size_t)N * 96;      // 7.7 MB
  float* Wp = ws + off; off += (size_t)HPAD * D_DIM;
  float* st = ws + off; off += 512;                 // sums1,sqs1,sums2,sqs2
  float* sc = ws + off; off += 512;                 // scale1,shift1,scale2,shift2
  float* Ge = H;  // H is dead after GEMM1 -> reuse for G_e

  const float invE = 1.0f / (float)E;

  k_zero<<<2, 256, 0, stream>>>(st, 512);
  k_padw1<<<(HPAD * D_DIM + 255) / 256, 256, 0, stream>>>(W1, Wp);

  k_proj<<<N / 32, 192, 0, stream>>>(h_V, W_va, b_va, Vl);            // V_local [N,96]
  k_proj<<<E / 32, 192, 0, stream>>>(h_E, W_vd, b_vd, Ve);            // V_edge  [E,96]

  k_geom<<<(E * 64 + 255) / 256, 256, 0, stream>>>(Ve, Vl, rot, trans, rbf, eidx, H, E);

  k_gemm1<<<E / 32, 256, 0, stream>>>(H, Wp, b1, Y1);                 // H @ W1 + b1
  k_stats<<<E / 256, 256, 0, stream>>>(Y1, st, st + 128, 256);
  k_scale_shift<<<1, 128, 0, stream>>>(st, st + 128, g1, be1, sc, sc + 128, invE);

  k_gemm_bn<<<E / 64, 256, 0, stream>>>(Y1, sc, sc + 128, W2, b2, Y2);  // relu(bn)@W2+b2
  k_stats<<<E / 256, 256, 0, stream>>>(Y2, st + 256, st + 384, 256);
  k_scale_shift<<<1, 128, 0, stream>>>(st + 256, st + 384, g2, be2, sc + 256, sc + 384, invE);

  k_gemm_bn<<<E / 64, 256, 0, stream>>>(Y2, sc + 256, sc + 384, W3, b3, Ge);  // G_e

  k_gemm4<<<E / 32, 256, 0, stream>>>(h_E, Ge, W_m, b_m, (float*)d_out);
}